// NodeAttentionHead_75642964017820
// MI455X (gfx1250) — compile-verified
//
#include <hip/hip_runtime.h>
#include <math.h>

#define N_NODES   50000
#define NODE_IN   128
#define NODE_OUT  64
#define EDGE_IN   32
#define N_EDGES   400000
#define E2        (2 * N_EDGES)
#define ALPHA     0.2f
#define EPS_F     1e-12f

// ---- workspace layout (in floats) ----
#define OFF_MSG     0                         // N_NODES*64
#define OFF_ATTSUM  (N_NODES * 64)            // N_NODES
#define OFF_VAR     (OFF_ATTSUM + N_NODES)    // 4 floats = 2 doubles (8B aligned: byte off 13,000,000)
#define OFF_S0      (OFF_VAR + 4)             // N_NODES
#define OFF_S1      (OFF_S0 + N_NODES)        // N_NODES
#define OFF_HN      (OFF_S1 + N_NODES)        // N_NODES
#define OFF_ED      (OFF_HN + N_NODES)        // N_EDGES
#define OFF_Z       (OFF_ED + N_EDGES)        // E2
#define ZERO_FLOATS (OFF_VAR + 4)             // zero msg + att_sum + var accumulators

typedef float v2f __attribute__((ext_vector_type(2)));
typedef float v8f __attribute__((ext_vector_type(8)));

__device__ __forceinline__ float atomic_add_f32(float* p, float v) {
  return __hip_atomic_fetch_add(p, v, __ATOMIC_RELAXED, __HIP_MEMORY_SCOPE_AGENT);
}
__device__ __forceinline__ double atomic_add_f64(double* p, double v) {
  return __hip_atomic_fetch_add(p, v, __ATOMIC_RELAXED, __HIP_MEMORY_SCOPE_AGENT);
}

// ---------------- K0: zero scratch regions ----------------
__global__ void k_zero(float* __restrict__ p, int n) {
  int i = blockIdx.x * blockDim.x + threadIdx.x;
  if (i < n) p[i] = 0.0f;
}

// ---------------- K1: h_v = X @ W + b via V_WMMA_F32_16X16X4_F32 ----------------
// block = 256 threads = 8 waves; each wave computes a 16x64 output tile (4 N-tiles).
// h_v written to d_out[row*128 + 0..63].
__global__ __launch_bounds__(256) void k_gemm_wmma(const float* __restrict__ X,
                                                   const float* __restrict__ Wg,
                                                   const float* __restrict__ bias,
                                                   float* __restrict__ out) {
  __shared__ float sWt[NODE_OUT * NODE_IN]; // transposed: sWt[n*128 + k] = W[k][n]
  __shared__ float sb[NODE_OUT];
  for (int idx = threadIdx.x; idx < NODE_IN * NODE_OUT; idx += 256) {
    int k = idx >> 6, n = idx & 63;
    sWt[n * NODE_IN + k] = Wg[idx];
  }
  if (threadIdx.x < NODE_OUT) sb[threadIdx.x] = bias[threadIdx.x];
  __syncthreads();

  const int wave = threadIdx.x >> 5;
  const int lane = threadIdx.x & 31;
  const int m    = lane & 15;            // row within 16-row tile (A) / col within 16-col tile (B)
  const int kOff = (lane & 16) ? 2 : 0;  // K sub-offset per ISA A/B fragment layout
  const int row0 = blockIdx.x * 128 + wave * 16;
  int row = row0 + m;
  int rowC = row < N_NODES ? row : (N_NODES - 1);
  const float* arow = X + (size_t)rowC * NODE_IN;

  v8f acc0 = {}, acc1 = {}, acc2 = {}, acc3 = {};
  for (int k0 = 0; k0 < NODE_IN; k0 += 4) {
    v2f a = *(const v2f*)(arow + k0 + kOff);
    v2f b0 = *(const v2f*)(&sWt[(0 * 16 + m) * NODE_IN + k0 + kOff]);
    v2f b1 = *(const v2f*)(&sWt[(1 * 16 + m) * NODE_IN + k0 + kOff]);
    v2f b2 = *(const v2f*)(&sWt[(2 * 16 + m) * NODE_IN + k0 + kOff]);
    v2f b3 = *(const v2f*)(&sWt[(3 * 16 + m) * NODE_IN + k0 + kOff]);
    acc0 = __builtin_amdgcn_wmma_f32_16x16x4_f32(false, a, false, b0, (short)0, acc0, false, false);
    acc1 = __builtin_amdgcn_wmma_f32_16x16x4_f32(false, a, false, b1, (short)0, acc1, false, false);
    acc2 = __builtin_amdgcn_wmma_f32_16x16x4_f32(false, a, false, b2, (short)0, acc2, false, false);
    acc3 = __builtin_amdgcn_wmma_f32_16x16x4_f32(false, a, false, b3, (short)0, acc3, false, false);
  }

  const int mBase = (lane & 16) ? 8 : 0;   // C/D layout: lanes>=16 hold M=8..15
#pragma unroll
  for (int nt = 0; nt < 4; ++nt) {
    v8f acc = (nt == 0) ? acc0 : (nt == 1) ? acc1 : (nt == 2) ? acc2 : acc3;
    int n = nt * 16 + m;
    float bn = sb[n];
#pragma unroll
    for (int i = 0; i < 8; ++i) {
      int r = row0 + mBase + i;
      if (r < N_NODES) out[(size_t)r * 128 + n] = acc[i] + bn;
    }
  }
}

// ---------------- K2: per-node s0 = h·a0, s1 = h·a1, hn = ||h|| ----------------
__global__ __launch_bounds__(256) void k_node_stats(const float* __restrict__ out,
                                                    const float* __restrict__ a_node,
                                                    float* __restrict__ s0,
                                                    float* __restrict__ s1,
                                                    float* __restrict__ hn) {
  int wave = threadIdx.x >> 5, lane = threadIdx.x & 31;
  int v = blockIdx.x * 8 + wave;
  if (v >= N_NODES) return;
  const float* h = out + (size_t)v * 128;
  float h0 = h[lane], h1 = h[lane + 32];
  float p0 = h0 * a_node[lane]      + h1 * a_node[lane + 32];
  float p1 = h0 * a_node[64 + lane] + h1 * a_node[96 + lane];
  float sq = h0 * h0 + h1 * h1;
  for (int o = 16; o > 0; o >>= 1) {
    p0 += __shfl_xor(p0, o);
    p1 += __shfl_xor(p1, o);
    sq += __shfl_xor(sq, o);
  }
  if (lane == 0) { s0[v] = p0; s1[v] = p1; hn[v] = sqrtf(sq); }
}

// ---------------- K3: per-edge ed = edge_fts · a2 ----------------
__global__ __launch_bounds__(256) void k_edge_dot(const float* __restrict__ ef,
                                                  const float* __restrict__ a_node,
                                                  float* __restrict__ ed) {
  int wave = threadIdx.x >> 5, lane = threadIdx.x & 31;
  int j = blockIdx.x * 8 + wave;
  if (j >= N_EDGES) return;
  float p = ef[(size_t)j * EDGE_IN + lane] * a_node[128 + lane];
  for (int o = 16; o > 0; o >>= 1) p += __shfl_xor(p, o);
  if (lane == 0) ed[j] = p;
}

// ---------------- K4: attention logits + exp + segment-sum of att by src ----------------
__global__ __launch_bounds__(256) void k_att_pass1(const int* __restrict__ eflat,
                                                   const float* __restrict__ s0,
                                                   const float* __restrict__ s1,
                                                   const float* __restrict__ ed,
                                                   float* __restrict__ z,
                                                   float* __restrict__ att_sum) {
  int u = blockIdx.x * blockDim.x + threadIdx.x;
  if (u >= E2) return;
  int src, dst, j;
  if (u < N_EDGES) { j = u;            src = eflat[2 * j];     dst = eflat[2 * j + 1]; }
  else             { j = u - N_EDGES;  src = eflat[2 * j + 1]; dst = eflat[2 * j];     }
  float logit = s0[src] + s1[dst] + ed[j];
  float zz = (logit >= 0.0f) ? logit : ALPHA * logit;
  z[u] = zz;
  atomic_add_f32(&att_sum[src], __expf(zz) * 0.0f + expf(zz));
}

// ---------------- K5: message scatter + variance partials ----------------
// wave per directed edge; 8 edges per block.
__global__ __launch_bounds__(256) void k_scatter(const int* __restrict__ eflat,
                                                 const float* __restrict__ z,
                                                 const float* __restrict__ att_sum,
                                                 const float* __restrict__ out,
                                                 float* __restrict__ msg,
                                                 double* __restrict__ var_acc) {
  __shared__ float wv[8];
  int wave = threadIdx.x >> 5, lane = threadIdx.x & 31;
  int u = blockIdx.x * 8 + wave;
  float w = 0.0f;
  bool valid = (u < E2);
  if (valid) {
    int src, dst, j;
    if (u < N_EDGES) { j = u;           src = eflat[2 * j];     dst = eflat[2 * j + 1]; }
    else             { j = u - N_EDGES; src = eflat[2 * j + 1]; dst = eflat[2 * j];     }
    w = z[u] - logf(att_sum[src]);
    float hd0 = out[(size_t)dst * 128 + lane];
    float hd1 = out[(size_t)dst * 128 + lane + 32];
    atomic_add_f32(&msg[(size_t)src * 64 + lane],      hd0 * w);
    atomic_add_f32(&msg[(size_t)src * 64 + lane + 32], hd1 * w);
  }
  if (lane == 0) wv[wave] = valid ? w : 0.0f;
  __syncthreads();
  if (threadIdx.x == 0) {
    double s = 0.0, s2 = 0.0;
    for (int i = 0; i < 8; ++i) { double x = (double)wv[i]; s += x; s2 += x * x; }
    atomic_add_f64(&var_acc[0], s);
    atomic_add_f64(&var_acc[1], s2);
  }
}

// ---------------- K6: agg = msg / max(||msg||, eps) * ||h|| * scale ----------------
__global__ __launch_bounds__(256) void k_finalize(const float* __restrict__ msg,
                                                  const float* __restrict__ hn,
                                                  const float* __restrict__ scale,
                                                  float* __restrict__ out) {
  int wave = threadIdx.x >> 5, lane = threadIdx.x & 31;
  int v = blockIdx.x * 8 + wave;
  if (v >= N_NODES) return;
  float m0 = msg[(size_t)v * 64 + lane];
  float m1 = msg[(size_t)v * 64 + lane + 32];
  float sq = m0 * m0 + m1 * m1;
  for (int o = 16; o > 0; o >>= 1) sq += __shfl_xor(sq, o);
  float coef = hn[v] * scale[0] / fmaxf(sqrtf(sq), EPS_F);
  out[(size_t)v * 128 + 64 + lane] = m0 * coef;
  out[(size_t)v * 128 + 96 + lane] = m1 * coef;
}

// ---------------- K7: write att_var (ddof=1) ----------------
__global__ void k_var(const double* __restrict__ var_acc, float* __restrict__ out) {
  if (blockIdx.x == 0 && threadIdx.x == 0) {
    const double n = (double)E2;
    double s = var_acc[0], s2 = var_acc[1];
    double var = (s2 - s * s / n) / (n - 1.0);
    out[(size_t)N_NODES * 128] = (float)var;
  }
}

extern "C" void kernel_launch(void* const* d_in, const int* in_sizes, int n_in,
                              void* d_out, int out_size, void* d_ws, size_t ws_size,
                              hipStream_t stream) {
  (void)in_sizes; (void)n_in; (void)out_size; (void)ws_size;
  const float* node_fts = (const float*)d_in[0];
  const float* edge_fts = (const float*)d_in[1];
  const int*   eflat    = (const int*)d_in[2];
  const float* W        = (const float*)d_in[3];
  const float* b        = (const float*)d_in[4];
  const float* a_node   = (const float*)d_in[5];
  const float* scale    = (const float*)d_in[6];
  float* out = (float*)d_out;
  float* ws  = (float*)d_ws;

  float*  msg     = ws + OFF_MSG;
  float*  att_sum = ws + OFF_ATTSUM;
  double* var_acc = (double*)(ws + OFF_VAR);
  float*  s0      = ws + OFF_S0;
  float*  s1      = ws + OFF_S1;
  float*  hn      = ws + OFF_HN;
  float*  ed      = ws + OFF_ED;
  float*  z       = ws + OFF_Z;

  // K0: zero msg, att_sum, variance accumulators (every launch; ws not re-poisoned)
  k_zero<<<(ZERO_FLOATS + 255) / 256, 256, 0, stream>>>(ws, ZERO_FLOATS);

  // K1: WMMA fp32 GEMM -> d_out[:, 0:64]
  k_gemm_wmma<<<(N_NODES + 127) / 128, 256, 0, stream>>>(node_fts, W, b, out);

  // K2: per-node scalars
  k_node_stats<<<(N_NODES + 7) / 8, 256, 0, stream>>>(out, a_node, s0, s1, hn);

  // K3: per-edge feature dot
  k_edge_dot<<<(N_EDGES + 7) / 8, 256, 0, stream>>>(edge_fts, a_node, ed);

  // K4: logits, exp, segment sum
  k_att_pass1<<<(E2 + 255) / 256, 256, 0, stream>>>(eflat, s0, s1, ed, z, att_sum);

  // K5: message scatter + variance partials
  k_scatter<<<(E2 + 7) / 8, 256, 0, stream>>>(eflat, z, att_sum, out, msg, var_acc);

  // K6: normalize + scale -> d_out[:, 64:128]
  k_finalize<<<(N_NODES + 7) / 8, 256, 0, stream>>>(msg, hn, scale, out);

  // K7: att_var
  k_var<<<1, 1, 0, stream>>>(var_acc, out);
}